// ConvFunc_CGCNN_13194139533626
// MI455X (gfx1250) — compile-verified
//
#include <hip/hip_runtime.h>
#include <hip/hip_bf16.h>
#include <math.h>

#define NN 50000
#define NE 800000
#define FD 64
#define K3 192
#define BN_EPS 1e-5f

typedef __attribute__((ext_vector_type(2))) float v2f;
typedef __attribute__((ext_vector_type(8))) float v8f;

// LDS layout for transposed weights: wT[n*194 + k], stride 194 keeps b64
// fragment loads 8-byte aligned (n*194 + even) and bank-conflict friendly.
#define WSTRIDE 194
#define WTILE (FD * WSTRIDE)   // one matrix: 64*194 floats

// ws layout (floats):
//  [0)                 agg            NN*FD = 3,200,000
//  [3200000)           edge stats     256  (sum_m, sq_m, sum_g, sq_g)
//  [3200256)           edge bn params 256  (scale_m, shift_m, scale_g, shift_g)
//  [3200512)           node stats     128  (sum_n, sq_n)
//  [3200640)           node bn params 128  (scale_n, shift_n)
#define WS_AGG     0
#define WS_ESTAT   (NN * FD)
#define WS_EBNP    (WS_ESTAT + 256)
#define WS_NSTAT   (WS_EBNP + 256)
#define WS_NBNP    (WS_NSTAT + 128)
#define WS_TOTAL   (WS_NBNP + 128)

__global__ __launch_bounds__(256) void zero_ws_kernel(float* p, int n) {
    int i = blockIdx.x * 256 + threadIdx.x;
    if (i < n) p[i] = 0.0f;
}

// Core 16-edge tile GEMM: Y_m = (x_src + x_dst + e) @ W_mlpt[0:64], Y_g likewise with W_gate[0:64]
// A layout (f32 16x4): lane L -> M = L&15, K = kk + 2*(L>>4) + r
// B layout (f32 4x16): lane L -> N = L&15, K = kk + 2*(L>>4) + r
// C/D layout:          lane L, reg r -> M = r + 8*(L>>4), N = L&15
__device__ __forceinline__ void gemm_tile(
    const float* __restrict__ node_feats, const float* __restrict__ edge_feats,
    const int* __restrict__ src, const int* __restrict__ dst,
    const float* lds_wm, const float* lds_wg,
    int tile, int lane, v8f accM[4], v8f accG[4])
{
    const int lm = lane & 15;
    const int hi = lane >> 4;
    const int e = tile * 16 + lm;
    const float* baseA[3];
    baseA[0] = node_feats + src[e] * FD + 2 * hi;
    baseA[1] = node_feats + dst[e] * FD + 2 * hi;
    baseA[2] = edge_feats + e * FD + 2 * hi;

#pragma unroll
    for (int seg = 0; seg < 3; ++seg) {
        const float* pA = baseA[seg];
#pragma unroll 4
        for (int kk = 0; kk < FD; kk += 4) {
            v2f a = *(const v2f*)(pA + kk);
#pragma unroll
            for (int t = 0; t < 4; ++t) {
                const int n = t * 16 + lm;
                const int wo = n * WSTRIDE + kk + 2 * hi;   // even -> 8B aligned
                v2f bm = *(const v2f*)(lds_wm + wo);
                accM[t] = __builtin_amdgcn_wmma_f32_16x16x4_f32(
                    false, a, false, bm, (short)0, accM[t], false, false);
                v2f bg = *(const v2f*)(lds_wg + wo);
                accG[t] = __builtin_amdgcn_wmma_f32_16x16x4_f32(
                    false, a, false, bg, (short)0, accG[t], false, false);
            }
        }
    }
}

__device__ __forceinline__ void load_weights_to_lds(
    float* lds_w, const float* __restrict__ W_m, const float* __restrict__ W_g)
{
    const int tid = threadIdx.x;
    // transpose [192,64] row-major -> wT[n*194 + k] for both matrices
    for (int idx = tid; idx < 2 * K3 * FD; idx += 256) {
        const int m = idx / (K3 * FD);
        const int r = idx - m * (K3 * FD);
        const int k = r >> 6;          // r / 64
        const int n = r & 63;          // r % 64
        const float* W = m ? W_g : W_m;
        lds_w[m * WTILE + n * WSTRIDE + k] = W[k * FD + n];
    }
}

// PASS 1: accumulate per-column sum / sum-of-squares of both linear outputs.
// PASS 2: apply BN affine + activations, scatter-add into agg.
template <int PASS>
__global__ __launch_bounds__(256, 4) void edge_pass_kernel(
    const float* __restrict__ node_feats, const float* __restrict__ edge_feats,
    const int* __restrict__ src, const int* __restrict__ dst,
    const float* __restrict__ W_m, const float* __restrict__ W_g,
    float* __restrict__ estat,      // PASS 1 output: 256 floats
    const float* __restrict__ ebnp, // PASS 2 input: 256 floats
    float* __restrict__ agg)        // PASS 2 output: NN*FD floats
{
    __shared__ float lds_w[2 * WTILE];
    __shared__ float lds_stats[256];

    const int tid = threadIdx.x;
    load_weights_to_lds(lds_w, W_m, W_g);
    if (PASS == 1 && tid < 256) lds_stats[tid] = 0.0f;
    __syncthreads();

    const int wave = tid >> 5;
    const int lane = tid & 31;
    const int tile = blockIdx.x * 8 + wave;   // 50000 tiles of 16 edges
    const int lm = lane & 15;
    const int hi = lane >> 4;

    v8f accM[4] = {}, accG[4] = {};
    gemm_tile(node_feats, edge_feats, src, dst,
              lds_w, lds_w + WTILE, tile, lane, accM, accG);

    if (PASS == 1) {
#pragma unroll
        for (int t = 0; t < 4; ++t) {
            float sM = 0.f, qM = 0.f, sG = 0.f, qG = 0.f;
#pragma unroll
            for (int r = 0; r < 8; ++r) {
                const float ym = accM[t][r];
                const float yg = accG[t][r];
                sM += ym; qM += ym * ym;
                sG += yg; qG += yg * yg;
            }
            const int n = t * 16 + lm;
            atomicAdd(&lds_stats[n], sM);
            atomicAdd(&lds_stats[64 + n], qM);
            atomicAdd(&lds_stats[128 + n], sG);
            atomicAdd(&lds_stats[192 + n], qG);
        }
        __syncthreads();
        if (tid < 256) unsafeAtomicAdd(&estat[tid], lds_stats[tid]);
    } else {
        float scM[4], shM[4], scG[4], shG[4];
#pragma unroll
        for (int t = 0; t < 4; ++t) {
            const int n = t * 16 + lm;
            scM[t] = ebnp[n];       shM[t] = ebnp[64 + n];
            scG[t] = ebnp[128 + n]; shG[t] = ebnp[192 + n];
        }
        const int dbase = tile * 16 + 8 * hi;   // 8 edges this half covers
        int4 d0 = *(const int4*)(dst + dbase);
        int4 d1 = *(const int4*)(dst + dbase + 4);
        const int drow[8] = { d0.x, d0.y, d0.z, d0.w, d1.x, d1.y, d1.z, d1.w };
#pragma unroll
        for (int t = 0; t < 4; ++t) {
            const int n = t * 16 + lm;
#pragma unroll
            for (int r = 0; r < 8; ++r) {
                const float ym = scM[t] * accM[t][r] + shM[t];
                const float hm = 1.0f / (1.0f + expf(-ym));
                const float yg = scG[t] * accG[t][r] + shG[t];
                const float hg = (yg > 0.0f) ? (yg + log1pf(expf(-yg)))
                                             : log1pf(expf(yg));
                unsafeAtomicAdd(&agg[drow[r] * FD + n], hm * hg);
            }
        }
    }
}

__global__ void finalize_edge_bn_kernel(
    const float* __restrict__ estat,
    const float* __restrict__ g_m, const float* __restrict__ be_m,
    const float* __restrict__ g_g, const float* __restrict__ be_g,
    float* __restrict__ ebnp)
{
    const int tid = threadIdx.x;   // 128 threads
    const int m = tid >> 6, n = tid & 63;
    const float inv_e = 1.0f / (float)NE;
    const float mu = estat[m * 128 + n] * inv_e;
    const float var = estat[m * 128 + 64 + n] * inv_e - mu * mu;
    const float inv = rsqrtf(var + BN_EPS);
    const float g = (m ? g_g : g_m)[n];
    const float be = (m ? be_g : be_m)[n];
    ebnp[m * 128 + n] = g * inv;
    ebnp[m * 128 + 64 + n] = be - mu * g * inv;
}

__global__ __launch_bounds__(256) void node_stats_kernel(
    const float* __restrict__ agg, float* __restrict__ nstat)
{
    __shared__ float ls[128];
    const int tid = threadIdx.x;
    if (tid < 128) ls[tid] = 0.0f;
    __syncthreads();
    const int n = tid & 63;
    float s = 0.f, q = 0.f;
    for (int i = blockIdx.x * 4 + (tid >> 6); i < NN; i += gridDim.x * 4) {
        const float v = agg[i * FD + n];
        s += v; q += v * v;
    }
    atomicAdd(&ls[n], s);
    atomicAdd(&ls[64 + n], q);
    __syncthreads();
    if (tid < 128) unsafeAtomicAdd(&nstat[tid], ls[tid]);
}

__global__ void finalize_node_bn_kernel(
    const float* __restrict__ nstat,
    const float* __restrict__ g_n, const float* __restrict__ be_n,
    float* __restrict__ nbnp)
{
    const int n = threadIdx.x;   // 64 threads
    const float inv_n = 1.0f / (float)NN;
    const float mu = nstat[n] * inv_n;
    const float var = nstat[64 + n] * inv_n - mu * mu;
    const float inv = rsqrtf(var + BN_EPS);
    nbnp[n] = g_n[n] * inv;
    nbnp[64 + n] = be_n[n] - mu * g_n[n] * inv;
}

__global__ __launch_bounds__(256) void node_out_kernel(
    const float* __restrict__ agg, const float* __restrict__ node_feats,
    const float* __restrict__ nbnp, float* __restrict__ out)
{
    const int idx = blockIdx.x * 256 + threadIdx.x;   // NN*FD = 3,200,000
    const int n = idx & 63;
    const float x = nbnp[n] * agg[idx] + nbnp[64 + n] + node_feats[idx];
    out[idx] = 1.0f / (1.0f + expf(-x));
}

__global__ __launch_bounds__(256) void copy_edge_kernel(
    const float4* __restrict__ src, float4* __restrict__ dst)
{
    const int idx = blockIdx.x * 256 + threadIdx.x;   // NE*FD/4 = 12,800,000
    dst[idx] = src[idx];
}

extern "C" void kernel_launch(void* const* d_in, const int* in_sizes, int n_in,
                              void* d_out, int out_size, void* d_ws, size_t ws_size,
                              hipStream_t stream) {
    const float* node_feats = (const float*)d_in[0];
    const float* edge_feats = (const float*)d_in[1];
    const int*   src        = (const int*)d_in[2];
    const int*   dst        = (const int*)d_in[3];
    const float* W_mlpt     = (const float*)d_in[4];
    // d_in[5] = b_mlpt: bias cancels inside BatchNorm (mean-subtracted) -> unused
    const float* g_mlpt     = (const float*)d_in[6];
    const float* be_mlpt    = (const float*)d_in[7];
    const float* W_gate     = (const float*)d_in[8];
    // d_in[9] = b_gate: likewise unused
    const float* g_gate     = (const float*)d_in[10];
    const float* be_gate    = (const float*)d_in[11];
    const float* g_node     = (const float*)d_in[12];
    const float* be_node    = (const float*)d_in[13];

    float* ws    = (float*)d_ws;
    float* agg   = ws + WS_AGG;
    float* estat = ws + WS_ESTAT;
    float* ebnp  = ws + WS_EBNP;
    float* nstat = ws + WS_NSTAT;
    float* nbnp  = ws + WS_NBNP;

    float* node_out = (float*)d_out;
    float* edge_out = (float*)d_out + (size_t)NN * FD;

    // 1. zero agg + stat accumulators
    zero_ws_kernel<<<(WS_TOTAL + 255) / 256, 256, 0, stream>>>(ws, WS_TOTAL);

    // 2. pass 1: WMMA GEMMs + column sum / sumsq for edge-BN statistics
    edge_pass_kernel<1><<<NE / 16 / 8, 256, 0, stream>>>(
        node_feats, edge_feats, src, dst, W_mlpt, W_gate, estat, nullptr, nullptr);

    // 3. edge BN scale/shift
    finalize_edge_bn_kernel<<<1, 128, 0, stream>>>(
        estat, g_mlpt, be_mlpt, g_gate, be_gate, ebnp);

    // 4. pass 2: recompute GEMMs, BN + sigmoid*softplus, scatter-add to agg
    edge_pass_kernel<2><<<NE / 16 / 8, 256, 0, stream>>>(
        node_feats, edge_feats, src, dst, W_mlpt, W_gate, nullptr, ebnp, agg);

    // 5-6. node BN statistics + scale/shift
    node_stats_kernel<<<256, 256, 0, stream>>>(agg, nstat);
    finalize_node_bn_kernel<<<1, 64, 0, stream>>>(nstat, g_node, be_node, nbnp);

    // 7. node output: sigmoid(BN(agg) + residual)
    node_out_kernel<<<(NN * FD) / 256, 256, 0, stream>>>(
        agg, node_feats, nbnp, node_out);

    // 8. pass-through edge features
    copy_edge_kernel<<<(NE * FD / 4) / 256, 256, 0, stream>>>(
        (const float4*)edge_feats, (float4*)edge_out);
}